// EvaM1Primal_40656160424183
// MI455X (gfx1250) — compile-verified
//
#include <hip/hip_runtime.h>
#include <hip/hip_bf16.h>
#include <math.h>

// Problem constants (from reference): B=16, N=1024, C=768, H=12, HD=64, m=N (1 mini-batch)
#define DIMC 768
#define NH   12
#define HDIM 64
#define SEQ  1024
#define NBAT 16
#define ROWS (NBAT * SEQ)      // 16384
#define QKV_COLS (3 * DIMC)    // 2304

typedef float v2f __attribute__((ext_vector_type(2)));
typedef float v8f __attribute__((ext_vector_type(8)));

static __device__ __forceinline__ v8f vzero8() {
    v8f z;
    #pragma unroll
    for (int i = 0; i < 8; ++i) z[i] = 0.0f;
    return z;
}

// CDNA5 fp32 tensor op: D(16x16) = A(16x4) * B(4x16) + C
static __device__ __forceinline__ v8f wmma_k4(v8f c, v2f a, v2f b) {
    return __builtin_amdgcn_wmma_f32_16x16x4_f32(
        /*neg_a=*/false, a, /*neg_b=*/false, b,
        /*c_mod=*/(short)0, c, /*reuse_a=*/false, /*reuse_b=*/false);
}

static __device__ __forceinline__ float wave_sum(float v) {
    #pragma unroll
    for (int off = 16; off > 0; off >>= 1) v += __shfl_xor(v, off, 32);
    return v;
}

// ---- CDNA5 async global->LDS (ASYNCcnt-tracked), 16 bytes per lane ---------
static __device__ __forceinline__ void async_b128(const float* g, const float* lds) {
    const unsigned loff = (unsigned)(uintptr_t)lds;  // generic shared ptr: low 32b = LDS addr
    asm volatile("global_load_async_to_lds_b128 %0, %1, off"
                 :: "v"(loff), "v"(g) : "memory");
}
#define WAIT_ASYNC(n) asm volatile("s_wait_asynccnt %0" :: "i"(n) : "memory")

// Async-copy a 64x64 f32 tile (row stride `ld` floats in global) into padded LDS.
// 256 threads x 4 ops, each op = one row-aligned 4-float chunk (16B aligned).
static __device__ __forceinline__ void async_tile64(const float* gbase, int ld,
                                                    float (*dst)[68], int tid) {
    #pragma unroll
    for (int i = 0; i < 4; ++i) {
        const int cid = tid + 256 * i;            // 0..1023
        const int rr = cid >> 4, c4 = (cid & 15) << 2;
        async_b128(gbase + (size_t)rr * ld + c4, &dst[rr][c4]);
    }
}

// C(16x16) += A[r0+..][k] * B[k][c0+..] with A row-major [M][K], B row-major [K][N]
static __device__ __forceinline__ void mma_tile(v8f& acc, const float (*As)[68],
                                                const float (*Bs)[68], int r0, int c0, int lane) {
    const int half = lane >> 4, l15 = lane & 15;
    #pragma unroll
    for (int kk = 0; kk < 64; kk += 4) {
        v2f a, b;
        a[0] = As[r0 + l15][kk + 2 * half];
        a[1] = As[r0 + l15][kk + 2 * half + 1];
        b[0] = Bs[kk + 2 * half][c0 + l15];
        b[1] = Bs[kk + 2 * half + 1][c0 + l15];
        acc = wmma_k4(acc, a, b);
    }
}

// Same but B is stored transposed [N][K] (weight-natural layout, async-copyable)
static __device__ __forceinline__ void mma_tile_nt(v8f& acc, const float (*As)[68],
                                                   const float (*BsT)[68], int r0, int c0, int lane) {
    const int half = lane >> 4, l15 = lane & 15;
    #pragma unroll
    for (int kk = 0; kk < 64; kk += 4) {
        v2f a, b;
        a[0] = As[r0 + l15][kk + 2 * half];
        a[1] = As[r0 + l15][kk + 2 * half + 1];
        b[0] = BsT[c0 + l15][kk + 2 * half];
        b[1] = BsT[c0 + l15][kk + 2 * half + 1];
        acc = wmma_k4(acc, a, b);
    }
}

// A read transposed: A[M][K] = KsT[K][M]  (grad_W1 = XK^T @ g)
static __device__ __forceinline__ void mma_tile_at(v8f& acc, const float (*KsT)[68],
                                                   const float (*Bs)[68], int m0, int c0, int lane) {
    const int half = lane >> 4, l15 = lane & 15;
    #pragma unroll
    for (int kk = 0; kk < 64; kk += 4) {
        v2f a, b;
        a[0] = KsT[kk + 2 * half][m0 + l15];
        a[1] = KsT[kk + 2 * half + 1][m0 + l15];
        b[0] = Bs[kk + 2 * half][c0 + l15];
        b[1] = Bs[kk + 2 * half + 1][c0 + l15];
        acc = wmma_k4(acc, a, b);
    }
}

// ---------------------------------------------------------------------------
// Kernel 1: QKV GEMM  [16384,768] x [768,2304]^T  -> q,k,v in [B,H,N,64]
// Double-buffered async-to-LDS pipeline.
// ---------------------------------------------------------------------------
__global__ __launch_bounds__(256) void qkv_gemm_kernel(
    const float* __restrict__ x, const float* __restrict__ w,
    const float* __restrict__ qbias, const float* __restrict__ vbias,
    float* __restrict__ qo, float* __restrict__ ko, float* __restrict__ vo) {
    __shared__ float As[2][64][68];
    __shared__ float Bs[2][64][68];   // [N][K] layout
    const int tid = threadIdx.x;
    const int lane = tid & 31, wv = tid >> 5;
    const int half = lane >> 4, l15 = lane & 15;
    const int row0 = blockIdx.x * 64, col0 = blockIdx.y * 64;

    v8f acc[2];
    acc[0] = vzero8(); acc[1] = vzero8();

    const float* xb = x + (size_t)row0 * DIMC;
    const float* wb = w + (size_t)col0 * DIMC;
    async_tile64(xb, DIMC, As[0], tid);
    async_tile64(wb, DIMC, Bs[0], tid);

    int buf = 0;
    for (int kb = 0; kb < DIMC; kb += 64, buf ^= 1) {
        if (kb + 64 < DIMC) {
            async_tile64(xb + kb + 64, DIMC, As[buf ^ 1], tid);
            async_tile64(wb + kb + 64, DIMC, Bs[buf ^ 1], tid);
            WAIT_ASYNC(8);      // 8 new in flight; previous 8 (current buf) complete in order
        } else {
            WAIT_ASYNC(0);
        }
        __syncthreads();
        #pragma unroll
        for (int t = 0; t < 2; ++t) {
            const int tl = wv * 2 + t;
            mma_tile_nt(acc[t], As[buf], Bs[buf], (tl & 3) * 16, (tl >> 2) * 16, lane);
        }
        __syncthreads();
    }

    #pragma unroll
    for (int t = 0; t < 2; ++t) {
        const int tl = wv * 2 + t, r0 = (tl & 3) * 16, c0 = (tl >> 2) * 16;
        #pragma unroll
        for (int r = 0; r < 8; ++r) {
            const int grow = row0 + r0 + r + 8 * half;
            const int gcol = col0 + c0 + l15;
            const int which = gcol / DIMC, rem = gcol % DIMC;
            const int hh = rem >> 6, dd = rem & 63;
            const float bias = (which == 0) ? qbias[rem] : (which == 2 ? vbias[rem] : 0.0f);
            const int bi = grow >> 10, nn = grow & 1023;
            float* dst = (which == 0) ? qo : (which == 1 ? ko : vo);
            dst[(((size_t)bi * NH + hh) * SEQ + nn) * HDIM + dd] = acc[t][r] + bias;
        }
    }
}

// ---------------------------------------------------------------------------
// Kernel 2: per-token learning rate  eta[b,h,n] = sigmoid(x.lrw[h]+lrb[h])/64
// ---------------------------------------------------------------------------
__global__ __launch_bounds__(256) void lr_eta_kernel(
    const float* __restrict__ x, const float* __restrict__ lrw,
    const float* __restrict__ lrb, float* __restrict__ eta) {
    __shared__ float wsm[NH][DIMC];
    const int tid = threadIdx.x, lane = tid & 31, wv = tid >> 5;
    // async-stage lr weights: 9216 floats = 2304 x b128 = 9 ops/thread
    float* wflat = &wsm[0][0];
    #pragma unroll
    for (int j = 0; j < 9; ++j) {
        const int i = tid * 4 + j * 1024;
        async_b128(lrw + i, wflat + i);
    }
    WAIT_ASYNC(0);
    __syncthreads();

    const int pair = blockIdx.x * 8 + wv;  // token index in [0, 16384)
    float p[NH];
    #pragma unroll
    for (int h = 0; h < NH; ++h) p[h] = 0.0f;
    const float* xr = x + (size_t)pair * DIMC;
    for (int c = lane; c < DIMC; c += 32) {
        const float xv = xr[c];
        #pragma unroll
        for (int h = 0; h < NH; ++h) p[h] += xv * wsm[h][c];
    }
    #pragma unroll
    for (int h = 0; h < NH; ++h) p[h] = wave_sum(p[h]);
    if (lane == 0) {
        const int bi = pair >> 10, nn = pair & 1023;
        for (int h = 0; h < NH; ++h) {
            const float t = p[h] + lrb[h];
            const float s = 1.0f / (1.0f + expf(-t));
            eta[((size_t)bi * NH + h) * SEQ + nn] = s * (1.0f / (float)HDIM);
        }
    }
}

// ---------------------------------------------------------------------------
// Kernel 3: TTT single mini-batch, one workgroup per (b,h)
// ---------------------------------------------------------------------------
__global__ __launch_bounds__(256) void ttt_kernel(
    const float* __restrict__ q, const float* __restrict__ k, const float* __restrict__ v,
    const float* __restrict__ eta, const float* __restrict__ nw, const float* __restrict__ nb,
    const float* __restrict__ W1, const float* __restrict__ b1, float* __restrict__ hidden) {
    __shared__ float W1s[64][68];  // W1, later overwritten by W1n
    __shared__ float Ks[64][68];   // XK chunk (phase A) / XQ chunk (phase B)
    __shared__ float Zs[64][68];   // Z1, overwritten in place by g
    __shared__ float b1s[64], gms[64], bts[64];

    const int tid = threadIdx.x, lane = tid & 31, wv = tid >> 5;
    const int half = lane >> 4, l15 = lane & 15;
    const int bh = blockIdx.x, bi = bh / NH, hh = bh % NH;
    const float* qh = q + (size_t)bh * SEQ * HDIM;
    const float* kh = k + (size_t)bh * SEQ * HDIM;
    const float* vh = v + (size_t)bh * SEQ * HDIM;
    const float* ehp = eta + (size_t)bh * SEQ;

    async_tile64(W1 + (size_t)hh * HDIM * HDIM, HDIM, W1s, tid);
    if (tid < 64) {
        b1s[tid] = b1[hh * HDIM + tid];
        gms[tid] = nw[hh * HDIM + tid];
        bts[tid] = nb[hh * HDIM + tid];
    }
    WAIT_ASYNC(0);
    __syncthreads();

    v8f gacc[2];
    gacc[0] = vzero8(); gacc[1] = vzero8();
    float gb = 0.0f;

    // -------- Phase A: accumulate grad_W1, grad_b1 over 16 chunks of 64 rows
    for (int ch = 0; ch < 16; ++ch) {
        const int rowbase = ch * 64;
        async_tile64(kh + (size_t)rowbase * HDIM, HDIM, Ks, tid);
        WAIT_ASYNC(0);
        __syncthreads();

        // Z1 = XK @ W1 + b1
        #pragma unroll
        for (int t = 0; t < 2; ++t) {
            const int tl = wv * 2 + t, r0 = (tl & 3) * 16, c0 = (tl >> 2) * 16;
            v8f z = vzero8();
            mma_tile(z, Ks, W1s, r0, c0, lane);
            #pragma unroll
            for (int r = 0; r < 8; ++r)
                Zs[r0 + r + 8 * half][c0 + l15] = z[r] + b1s[c0 + l15];
        }
        __syncthreads();

        // fused LN-L2 backward, scaled by eta/m, in place Zs := g
        for (int rw = wv; rw < 64; rw += 8) {
            const int grow = rowbase + rw;
            const float z0 = Zs[rw][lane], z1 = Zs[rw][lane + 32];
            const float mu = wave_sum(z0 + z1) * (1.0f / 64.0f);
            const float d0 = z0 - mu, d1 = z1 - mu;
            const float var = wave_sum(d0 * d0 + d1 * d1) * (1.0f / 64.0f);
            const float rstd = rsqrtf(var + 1e-6f);
            const float xh0 = d0 * rstd, xh1 = d1 * rstd;
            const float gm0 = gms[lane], gm1 = gms[lane + 32];
            const float tgt0 = vh[(size_t)grow * HDIM + lane]      - Ks[rw][lane];
            const float tgt1 = vh[(size_t)grow * HDIM + lane + 32] - Ks[rw][lane + 32];
            const float g0 = (gm0 * xh0 + bts[lane]      - tgt0) * gm0;
            const float g1 = (gm1 * xh1 + bts[lane + 32] - tgt1) * gm1;
            const float sg  = wave_sum(g0 + g1);
            const float sgx = wave_sum(g0 * xh0 + g1 * xh1);
            const float sc = rstd * (1.0f / 64.0f) * (ehp[grow] * (1.0f / (float)SEQ));
            Zs[rw][lane]      = (64.0f * g0 - sg - xh0 * sgx) * sc;
            Zs[rw][lane + 32] = (64.0f * g1 - sg - xh1 * sgx) * sc;
        }
        __syncthreads();

        if (tid < 64) {
            #pragma unroll 8
            for (int r = 0; r < 64; ++r) gb += Zs[r][tid];
        }
        // grad_W1 += XK^T @ g   (A transposed read from LDS)
        #pragma unroll
        for (int t = 0; t < 2; ++t) {
            const int tl = wv * 2 + t;
            mma_tile_at(gacc[t], Ks, Zs, (tl & 3) * 16, (tl >> 2) * 16, lane);
        }
        __syncthreads();
    }

    // W1n = W1 - grad_W1 (in place), b1n = b1 - grad_b1
    #pragma unroll
    for (int t = 0; t < 2; ++t) {
        const int tl = wv * 2 + t, r0 = (tl & 3) * 16, c0 = (tl >> 2) * 16;
        #pragma unroll
        for (int r = 0; r < 8; ++r) {
            const int dr = r0 + r + 8 * half, ec = c0 + l15;
            W1s[dr][ec] = W1s[dr][ec] - gacc[t][r];
        }
    }
    if (tid < 64) b1s[tid] -= gb;
    __syncthreads();

    // -------- Phase B: out = XQ + ln_fwd(XQ @ W1n + b1n), into [B,N,C]
    for (int ch = 0; ch < 16; ++ch) {
        const int rowbase = ch * 64;
        async_tile64(qh + (size_t)rowbase * HDIM, HDIM, Ks, tid);
        WAIT_ASYNC(0);
        __syncthreads();
        #pragma unroll
        for (int t = 0; t < 2; ++t) {
            const int tl = wv * 2 + t, r0 = (tl & 3) * 16, c0 = (tl >> 2) * 16;
            v8f z = vzero8();
            mma_tile(z, Ks, W1s, r0, c0, lane);
            #pragma unroll
            for (int r = 0; r < 8; ++r)
                Zs[r0 + r + 8 * half][c0 + l15] = z[r] + b1s[c0 + l15];
        }
        __syncthreads();
        for (int rw = wv; rw < 64; rw += 8) {
            const int grow = rowbase + rw;
            const float z0 = Zs[rw][lane], z1 = Zs[rw][lane + 32];
            const float mu = wave_sum(z0 + z1) * (1.0f / 64.0f);
            const float d0 = z0 - mu, d1 = z1 - mu;
            const float var = wave_sum(d0 * d0 + d1 * d1) * (1.0f / 64.0f);
            const float rstd = rsqrtf(var + 1e-6f);
            const float y0 = gms[lane]      * (d0 * rstd) + bts[lane];
            const float y1 = gms[lane + 32] * (d1 * rstd) + bts[lane + 32];
            const size_t ob = ((size_t)bi * SEQ + grow) * DIMC + hh * HDIM;
            hidden[ob + lane]      = Ks[rw][lane] + y0;
            hidden[ob + lane + 32] = Ks[rw][lane + 32] + y1;
        }
        __syncthreads();
    }
}

// ---------------------------------------------------------------------------
// Kernel 4: output projection  [16384,768] x [768,768]^T + bias
// ---------------------------------------------------------------------------
__global__ __launch_bounds__(256) void proj_gemm_kernel(
    const float* __restrict__ hidden, const float* __restrict__ w,
    const float* __restrict__ pbias, float* __restrict__ out) {
    __shared__ float As[2][64][68];
    __shared__ float Bs[2][64][68];   // [N][K] layout
    const int tid = threadIdx.x;
    const int lane = tid & 31, wv = tid >> 5;
    const int half = lane >> 4, l15 = lane & 15;
    const int row0 = blockIdx.x * 64, col0 = blockIdx.y * 64;

    v8f acc[2];
    acc[0] = vzero8(); acc[1] = vzero8();

    const float* hb = hidden + (size_t)row0 * DIMC;
    const float* wb = w + (size_t)col0 * DIMC;
    async_tile64(hb, DIMC, As[0], tid);
    async_tile64(wb, DIMC, Bs[0], tid);

    int buf = 0;
    for (int kb = 0; kb < DIMC; kb += 64, buf ^= 1) {
        if (kb + 64 < DIMC) {
            async_tile64(hb + kb + 64, DIMC, As[buf ^ 1], tid);
            async_tile64(wb + kb + 64, DIMC, Bs[buf ^ 1], tid);
            WAIT_ASYNC(8);
        } else {
            WAIT_ASYNC(0);
        }
        __syncthreads();
        #pragma unroll
        for (int t = 0; t < 2; ++t) {
            const int tl = wv * 2 + t;
            mma_tile_nt(acc[t], As[buf], Bs[buf], (tl & 3) * 16, (tl >> 2) * 16, lane);
        }
        __syncthreads();
    }

    #pragma unroll
    for (int t = 0; t < 2; ++t) {
        const int tl = wv * 2 + t, r0 = (tl & 3) * 16, c0 = (tl >> 2) * 16;
        #pragma unroll
        for (int r = 0; r < 8; ++r) {
            const int grow = row0 + r0 + r + 8 * half;
            const int gcol = col0 + c0 + l15;
            out[(size_t)grow * DIMC + gcol] = acc[t][r] + pbias[gcol];
        }
    }
}

// ---------------------------------------------------------------------------
extern "C" void kernel_launch(void* const* d_in, const int* in_sizes, int n_in,
                              void* d_out, int out_size, void* d_ws, size_t ws_size,
                              hipStream_t stream) {
    const float* x      = (const float*)d_in[0];
    const float* qkv_w  = (const float*)d_in[1];
    const float* q_bias = (const float*)d_in[2];
    const float* v_bias = (const float*)d_in[3];
    const float* proj_w = (const float*)d_in[4];
    const float* proj_b = (const float*)d_in[5];
    const float* lrw    = (const float*)d_in[6];
    const float* lrb    = (const float*)d_in[7];
    const float* nw     = (const float*)d_in[8];
    const float* nb     = (const float*)d_in[9];
    const float* W1     = (const float*)d_in[10];
    const float* b1     = (const float*)d_in[11];
    float* out = (float*)d_out;

    // workspace layout (floats): q,k,v [B,H,N,64] + eta [B,H,N] + hidden [B,N,C]
    float* ws = (float*)d_ws;
    const size_t QS = (size_t)NBAT * NH * SEQ * HDIM;  // 12,582,912
    float* qbuf   = ws;
    float* kbuf   = ws + QS;
    float* vbuf   = ws + 2 * QS;
    float* etabuf = ws + 3 * QS;
    float* hidden = etabuf + (size_t)NBAT * NH * SEQ;

    qkv_gemm_kernel<<<dim3(ROWS / 64, QKV_COLS / 64), 256, 0, stream>>>(
        x, qkv_w, q_bias, v_bias, qbuf, kbuf, vbuf);
    lr_eta_kernel<<<dim3(ROWS / 8), 256, 0, stream>>>(x, lrw, lrb, etabuf);
    ttt_kernel<<<dim3(NBAT * NH), 256, 0, stream>>>(
        qbuf, kbuf, vbuf, etabuf, nw, nb, W1, b1, hidden);
    proj_gemm_kernel<<<dim3(ROWS / 64, DIMC / 64), 256, 0, stream>>>(
        hidden, proj_w, proj_b, out);
}